// SSKernelNPLR_37993280700533
// MI455X (gfx1250) — compile-verified
//
#include <hip/hip_runtime.h>
#include <hip/hip_bf16.h>
#include <math.h>

// Problem constants (from reference)
#define H_   256
#define N_   32
#define M_   64       // 2N conjugate-extended state
#define L_   8192
#define LH_  4096     // L/2
#define LF_  4097     // L/2+1

typedef float v2f __attribute__((ext_vector_type(2)));
typedef float v8f __attribute__((ext_vector_type(8)));

// ---------------- tiny complex helpers (POD, no HIP vector-type quirks) ----
struct cplx { float x, y; };
__device__ __forceinline__ cplx mkc(float a, float b){ cplx c; c.x=a; c.y=b; return c; }
__device__ __forceinline__ cplx cadd(cplx a, cplx b){ return mkc(a.x+b.x, a.y+b.y); }
__device__ __forceinline__ cplx csub(cplx a, cplx b){ return mkc(a.x-b.x, a.y-b.y); }
__device__ __forceinline__ cplx cmul(cplx a, cplx b){ return mkc(a.x*b.x - a.y*b.y, a.x*b.y + a.y*b.x); }
__device__ __forceinline__ cplx cconj(cplx a){ return mkc(a.x, -a.y); }
__device__ __forceinline__ cplx cscale(cplx a, float s){ return mkc(a.x*s, a.y*s); }
__device__ __forceinline__ cplx crecip(cplx a){ float d = a.x*a.x + a.y*a.y; float r = 1.0f/d; return mkc(a.x*r, -a.y*r); }
__device__ __forceinline__ cplx cdiv(cplx a, cplx b){ return cmul(a, crecip(b)); }

// =====================================================================
// Kernel 1: discretization + closed-form dA construction.
// dA[h][n][b] = Dc[n]*(Ec[n]*delta(n,b) - Pc2[n]*(Qc2[b] + s[b]))
// with s[b] = Rc[b]*Ec[b] - (sum_m Rc[m]*Pc2[m]) * Qc2[b]
// grid: H blocks, 64 threads.
// =====================================================================
__global__ void setup_kernel(const float* __restrict__ log_dt,
                             const float* __restrict__ inv_w_real,
                             const float* __restrict__ w_imag,
                             const float* __restrict__ Pg,
                             float* __restrict__ dAr, float* __restrict__ dAi,
                             float* __restrict__ dt_out, cplx* __restrict__ wdt_out)
{
    __shared__ cplx D2[M_], E2[M_], P2[M_], Q2[M_], R2[M_], colf[M_];
    __shared__ float red[N_];
    __shared__ float sRm;
    __shared__ cplx sSRP;

    int h = blockIdx.x;
    int t = threadIdx.x;           // 0..63
    float dt = expf(log_dt[h]);
    if (t == 0) dt_out[h] = dt;

    if (t < N_) {
        int n = t;
        float wr = -expf(inv_w_real[h*N_ + n]);
        float wi = w_imag[h*N_ + n];
        cplx p = mkc(Pg[(h*N_ + n)*2 + 0], Pg[(h*N_ + n)*2 + 1]);
        cplx D = crecip(mkc(2.0f/dt - wr, -wi));     // 1/(2/dt - w)
        cplx E = mkc(2.0f/dt + wr, wi);              // 2/dt + w
        red[n] = (p.x*p.x + p.y*p.y) * D.x;          // |P|^2 * Re(D)
        wdt_out[h*N_ + n] = mkc(wr*dt, wi*dt);
        D2[n] = D; D2[n+N_] = cconj(D);
        E2[n] = E; E2[n+N_] = cconj(E);
        P2[n] = p; P2[n+N_] = cconj(p);
        Q2[n] = cconj(p); Q2[n+N_] = p;
    }
    __syncthreads();
    if (t == 0) {
        float s = 0.f;
        for (int i = 0; i < N_; ++i) s += red[i];
        sRm = 1.0f + 2.0f*s;
    }
    __syncthreads();
    if (t < N_) {
        int n = t;
        cplx rs = cscale(cmul(Q2[n], D2[n]), 1.0f/sRm);   // Rs = conj(P)*D / Rm
        R2[n] = rs; R2[n+N_] = cconj(rs);
    }
    __syncthreads();
    if (t == 0) {
        cplx s = mkc(0.f, 0.f);
        for (int m = 0; m < M_; ++m) s = cadd(s, cmul(R2[m], P2[m]));
        sSRP = s;
    }
    __syncthreads();
    {   // per-column factor: colf[b] = Qc2[b] + s[b]
        int b = t;
        cplx sb = csub(cmul(R2[b], E2[b]), cmul(sSRP, Q2[b]));
        colf[b] = cadd(Q2[b], sb);
    }
    __syncthreads();
    {   // row n = t: fill 64 entries
        int n = t;
        cplx Dn = D2[n];
        cplx DP = cmul(Dn, P2[n]);
        cplx DE = cmul(Dn, E2[n]);
        size_t base = (size_t)h*M_*M_ + (size_t)n*M_;
        for (int b = 0; b < M_; ++b) {
            cplx v = cmul(DP, colf[b]);
            v = mkc(-v.x, -v.y);
            if (b == n) v = cadd(v, DE);
            dAr[base + b] = v.x;
            dAi[base + b] = v.y;
        }
    }
}

// =====================================================================
// Kernel 2: complex 64x64 matrix squaring via V_WMMA_F32_16X16X4_F32.
// One block per h (512 threads = 16 waves), each wave owns one 16x16
// output tile. Cr = Ar*Ar - Ai*Ai ; Ci = Ar*Ai + Ai*Ar.
// 16 K-steps x 4 WMMA per wave.
// =====================================================================
__global__ void square_dA_kernel(const float* __restrict__ srcRe, const float* __restrict__ srcIm,
                                 float* __restrict__ dstRe, float* __restrict__ dstIm)
{
    __shared__ float Ar[M_*M_];
    __shared__ float Ai[M_*M_];
    int h = blockIdx.x;
    const float* sr = srcRe + (size_t)h*M_*M_;
    const float* si = srcIm + (size_t)h*M_*M_;
    for (int i = threadIdx.x; i < M_*M_; i += blockDim.x) { Ar[i] = sr[i]; Ai[i] = si[i]; }
    __syncthreads();

    int wv   = threadIdx.x >> 5;      // 0..15
    int lane = threadIdx.x & 31;
    int ti   = (wv >> 2) * 16;        // output row base
    int tj   = (wv & 3) * 16;         // output col base
    int hf   = lane >> 4;             // lane half
    int l15  = lane & 15;

    v8f accRR = {}, accII = {}, accRI = {}, accIR = {};
    int am = ti + l15;                // A fragment row
    int bn = tj + l15;                // B fragment col

#pragma unroll
    for (int k0 = 0; k0 < M_; k0 += 4) {
        int ak = k0 + 2*hf;           // K = 2*half + vgpr
        v2f aRe = { Ar[am*M_ + ak], Ar[am*M_ + ak + 1] };
        v2f aIm = { Ai[am*M_ + ak], Ai[am*M_ + ak + 1] };
        v2f bRe = { Ar[ak*M_ + bn], Ar[(ak+1)*M_ + bn] };
        v2f bIm = { Ai[ak*M_ + bn], Ai[(ak+1)*M_ + bn] };
        accRR = __builtin_amdgcn_wmma_f32_16x16x4_f32(false, aRe, false, bRe, (short)0, accRR, false, false);
        accII = __builtin_amdgcn_wmma_f32_16x16x4_f32(false, aIm, false, bIm, (short)0, accII, false, false);
        accRI = __builtin_amdgcn_wmma_f32_16x16x4_f32(false, aRe, false, bIm, (short)0, accRI, false, false);
        accIR = __builtin_amdgcn_wmma_f32_16x16x4_f32(false, aIm, false, bRe, (short)0, accIR, false, false);
    }

    float* dr = dstRe + (size_t)h*M_*M_;
    float* di = dstIm + (size_t)h*M_*M_;
#pragma unroll
    for (int r = 0; r < 8; ++r) {
        int m = ti + r + 8*hf;        // C/D layout: M = vgpr + 8*half
        int n = tj + l15;
        dr[m*M_ + n] = accRR[r] - accII[r];
        di[m*M_ + n] = accRI[r] + accIR[r];
    }
}

// =====================================================================
// Kernel 3: C~ = (C_ - dA_L^T C_)[:N], then the 4 Woodbury numerator
// vectors v* = {B,P} x {C~,Q} * dt.  grid: H blocks, 32 threads.
// =====================================================================
__global__ void ct_kernel(const float* __restrict__ dALr, const float* __restrict__ dALi,
                          const float* __restrict__ Bg, const float* __restrict__ Cg,
                          const float* __restrict__ Pg, const float* __restrict__ dt_arr,
                          cplx* __restrict__ vBC, cplx* __restrict__ vBQ,
                          cplx* __restrict__ vPC, cplx* __restrict__ vPQ)
{
    int h = blockIdx.x;
    int m = threadIdx.x;    // 0..31
    // prod[m] = sum_n dA_L[h][n][m] * Cext[n]
    cplx acc = mkc(0.f, 0.f);
    for (int n = 0; n < M_; ++n) {
        cplx cn;
        if (n < N_) cn = mkc(Cg[(h*N_ + n)*2], Cg[(h*N_ + n)*2 + 1]);
        else        cn = mkc(Cg[(h*N_ + n - N_)*2], -Cg[(h*N_ + n - N_)*2 + 1]);
        size_t idx = (size_t)h*M_*M_ + (size_t)n*M_ + m;
        cplx a = mkc(dALr[idx], dALi[idx]);
        acc = cadd(acc, cmul(a, cn));
    }
    cplx Cm = mkc(Cg[(h*N_ + m)*2], Cg[(h*N_ + m)*2 + 1]);
    cplx ct = csub(Cm, acc);
    float dt = dt_arr[h];
    cplx b = mkc(Bg[(h*N_ + m)*2], Bg[(h*N_ + m)*2 + 1]);
    cplx p = mkc(Pg[(h*N_ + m)*2], Pg[(h*N_ + m)*2 + 1]);
    cplx q = cconj(p);
    vBC[h*N_ + m] = cscale(cmul(b, ct), dt);
    vBQ[h*N_ + m] = cscale(cmul(b, q),  dt);
    vPC[h*N_ + m] = cscale(cmul(p, ct), dt);
    vPQ[h*N_ + m] = cscale(cmul(p, q),  dt);
}

// =====================================================================
// Kernel 4: Cauchy matvec (conjugate-pair symmetry) + rank-1 Woodbury.
// thread per (h, l).  grid: (ceil(LF/128), H), block 128.
// =====================================================================
__global__ void cauchy_kernel(const cplx* __restrict__ wdt,
                              const cplx* __restrict__ vBC, const cplx* __restrict__ vBQ,
                              const cplx* __restrict__ vPC, const cplx* __restrict__ vPQ,
                              cplx* __restrict__ kf)
{
    __shared__ cplx sw[N_], s00[N_], s01[N_], s10[N_], s11[N_];
    int h = blockIdx.y;
    int l = blockIdx.x * blockDim.x + threadIdx.x;
    if (threadIdx.x < N_) {
        int n = threadIdx.x;
        sw[n]  = wdt[h*N_ + n];
        s00[n] = vBC[h*N_ + n];
        s01[n] = vBQ[h*N_ + n];
        s10[n] = vPC[h*N_ + n];
        s11[n] = vPQ[h*N_ + n];
    }
    __syncthreads();
    if (l >= LF_) return;

    const float PI = 3.14159265358979323846f;
    float ang = -2.0f * PI * (float)l / (float)L_;
    float sn, cs; __sincosf(ang, &sn, &cs);
    cplx om  = mkc(cs, sn);
    cplx one = mkc(1.f, 0.f);
    cplx opo = cadd(one, om);                       // 1 + omega
    cplx z   = cdiv(cscale(csub(one, om), 2.0f), opo);

    cplx r00 = mkc(0,0), r01 = mkc(0,0), r10 = mkc(0,0), r11 = mkc(0,0);
    for (int n = 0; n < N_; ++n) {
        cplx w  = sw[n];
        cplx c1 = crecip(csub(z, w));               // 1/(z - w)
        cplx c2 = crecip(csub(z, cconj(w)));        // 1/(z - conj(w))
        r00 = cadd(r00, cadd(cmul(s00[n], c1), cmul(cconj(s00[n]), c2)));
        r01 = cadd(r01, cadd(cmul(s01[n], c1), cmul(cconj(s01[n]), c2)));
        r10 = cadd(r10, cadd(cmul(s10[n], c1), cmul(cconj(s10[n]), c2)));
        r11 = cadd(r11, cadd(cmul(s11[n], c1), cmul(cconj(s11[n]), c2)));
    }
    cplx kfv = csub(r00, cdiv(cmul(r01, r10), cadd(one, r11)));
    kfv = cmul(kfv, cdiv(mkc(2.f, 0.f), opo));      // * 2/(1+omega)
    kf[(size_t)h*LF_ + l] = kfv;
}

// =====================================================================
// Kernel 5: irfft(n=8192) via Hermitian half-size pack + 4096-pt inverse
// Stockham radix-2 FFT in LDS (ping-pong 64KB dynamic LDS; CDNA5 allows
// up to 320KB per workgroup). One block per h, 512 threads, 12 stages.
// =====================================================================
__global__ void irfft_kernel(const cplx* __restrict__ kf, float* __restrict__ out)
{
    extern __shared__ cplx smem[];          // 2 * 4096 cplx = 64KB
    cplx* b0 = smem;
    cplx* b1 = smem + LH_;
    int h = blockIdx.x;
    const cplx* K = kf + (size_t)h*LF_;
    const float PI = 3.14159265358979323846f;

    // pack: G[k] = E[k] + i*O[k],  E=(X[k]+conj(X[Lh-k]))/2,
    // O=(X[k]-conj(X[Lh-k]))/2 * e^{+2*pi*i*k/L}
    for (int k = threadIdx.x; k < LH_; k += blockDim.x) {
        cplx a  = K[k];
        cplx bq = cconj(K[LH_ - k]);
        cplx E  = cscale(cadd(a, bq), 0.5f);
        cplx Od = cscale(csub(a, bq), 0.5f);
        float ang = 2.0f * PI * (float)k / (float)L_;
        float sn, cs; __sincosf(ang, &sn, &cs);
        cplx O = cmul(Od, mkc(cs, sn));
        b0[k] = cadd(E, mkc(-O.y, O.x));    // E + i*O
    }
    __syncthreads();

    // inverse (sign=+1) Stockham radix-2, self-sorting, 12 stages
    cplx* src = b0;
    cplx* dst = b1;
    int l = LH_/2;          // 2048
    int mshift = 0;
    while (l >= 1) {
        int m = 1 << mshift;
        for (int idx = threadIdx.x; idx < LH_/2; idx += blockDim.x) {
            int j = idx >> mshift;
            int k = idx & (m - 1);
            cplx c0 = src[k + j*m];
            cplx c1 = src[k + j*m + (LH_/2)];   // l*m == LH_/2 always
            float ang = PI * (float)j / (float)l;
            float sn, cs; __sincosf(ang, &sn, &cs);
            cplx w = mkc(cs, sn);
            dst[k + 2*j*m]     = cadd(c0, c1);
            dst[k + 2*j*m + m] = cmul(w, csub(c0, c1));
        }
        __syncthreads();
        cplx* tmp = src; src = dst; dst = tmp;
        l >>= 1; mshift++;
    }
    // 12 stages -> result back in b0 (== src), natural order.
    float* o = out + (size_t)h*L_;
    const float scale = 1.0f / (float)LH_;
    for (int s = threadIdx.x; s < LH_; s += blockDim.x) {
        cplx z = src[s];
        o[2*s + 0] = z.x * scale;   // x[2s]   = Re
        o[2*s + 1] = z.y * scale;   // x[2s+1] = Im
    }
}

// =====================================================================
extern "C" void kernel_launch(void* const* d_in, const int* in_sizes, int n_in,
                              void* d_out, int out_size, void* d_ws, size_t ws_size,
                              hipStream_t stream)
{
    (void)in_sizes; (void)n_in; (void)out_size; (void)ws_size;
    const float* log_dt     = (const float*)d_in[0];
    const float* inv_w_real = (const float*)d_in[1];
    const float* w_imag     = (const float*)d_in[2];
    const float* Pg         = (const float*)d_in[3];
    const float* Bg         = (const float*)d_in[4];
    const float* Cg         = (const float*)d_in[5];
    float* out = (float*)d_out;

    // ---- workspace layout (~25.5 MB) ----
    char* ws = (char*)d_ws;
    size_t off = 0;
    auto alloc = [&](size_t bytes) -> void* {
        void* p = ws + off;
        off += (bytes + 255) & ~(size_t)255;
        return p;
    };
    const size_t matf = (size_t)H_ * M_ * M_ * sizeof(float);   // 4 MB
    float* dAr_a = (float*)alloc(matf);
    float* dAi_a = (float*)alloc(matf);
    float* dAr_b = (float*)alloc(matf);
    float* dAi_b = (float*)alloc(matf);
    float* dt_arr = (float*)alloc(H_ * sizeof(float));
    cplx*  wdt    = (cplx*)alloc((size_t)H_ * N_ * sizeof(cplx));
    cplx*  vBC    = (cplx*)alloc((size_t)H_ * N_ * sizeof(cplx));
    cplx*  vBQ    = (cplx*)alloc((size_t)H_ * N_ * sizeof(cplx));
    cplx*  vPC    = (cplx*)alloc((size_t)H_ * N_ * sizeof(cplx));
    cplx*  vPQ    = (cplx*)alloc((size_t)H_ * N_ * sizeof(cplx));
    cplx*  kf     = (cplx*)alloc((size_t)H_ * LF_ * sizeof(cplx));

    // 1) setup: discretize + build dA (into buffer a)
    setup_kernel<<<H_, 64, 0, stream>>>(log_dt, inv_w_real, w_imag, Pg,
                                        dAr_a, dAi_a, dt_arr, wdt);

    // 2) dA^8192 = 13 squarings (L is a power of two). Ping-pong a<->b;
    //    13 (odd) launches -> result lands in buffer b.
    for (int i = 0; i < 13; ++i) {
        const float* sr = (i & 1) ? dAr_b : dAr_a;
        const float* si = (i & 1) ? dAi_b : dAi_a;
        float* dr = (i & 1) ? dAr_a : dAr_b;
        float* di = (i & 1) ? dAi_a : dAi_b;
        square_dA_kernel<<<H_, 512, 0, stream>>>(sr, si, dr, di);
    }

    // 3) C~ and Woodbury numerator vectors
    ct_kernel<<<H_, N_, 0, stream>>>(dAr_b, dAi_b, Bg, Cg, Pg, dt_arr,
                                     vBC, vBQ, vPC, vPQ);

    // 4) Cauchy + Woodbury over 4097 FFT nodes
    dim3 cgrid((LF_ + 127) / 128, H_);
    cauchy_kernel<<<cgrid, 128, 0, stream>>>(wdt, vBC, vBQ, vPC, vPQ, kf);

    // 5) irfft (half-size packed inverse FFT), 64KB dynamic LDS
    irfft_kernel<<<H_, 512, 2 * LH_ * sizeof(cplx), stream>>>(kf, out);
}